// GraphSAGEFraudDetector_77687368450080
// MI455X (gfx1250) — compile-verified
//
#include <hip/hip_runtime.h>
#include <hip/hip_bf16.h>
#include <math.h>

#define HID 128

typedef __attribute__((ext_vector_type(2)))  float  v2f;
typedef __attribute__((ext_vector_type(8)))  float  v8f;

#define HAS_WMMA_F32X4 __has_builtin(__builtin_amdgcn_wmma_f32_16x16x4_f32)

// ---------------------------------------------------------------------------
// h0[i, 0:64]=x, [64:96]=user_emb, [96:112]=loc_emb, [112:128]=time@W_time+b
// one thread per (node, feature)
// ---------------------------------------------------------------------------
__global__ void encode_kernel(const float* __restrict__ x,
                              const int* __restrict__ user_ids,
                              const int* __restrict__ locations,
                              const float* __restrict__ timef,
                              const float* __restrict__ user_emb,
                              const float* __restrict__ loc_emb,
                              const float* __restrict__ W_time,
                              const float* __restrict__ b_time,
                              float* __restrict__ h0, int n)
{
    int t = blockIdx.x * blockDim.x + threadIdx.x;
    if (t >= n * HID) return;
    int node = t >> 7;
    int f = t & 127;
    float v;
    if (f < 64) {
        v = x[(size_t)node * 64 + f];
    } else if (f < 96) {
        v = user_emb[(size_t)user_ids[node] * 32 + (f - 64)];
    } else if (f < 112) {
        v = loc_emb[(size_t)locations[node] * 16 + (f - 96)];
    } else {
        int j = f - 112;
        v = b_time[j];
        #pragma unroll
        for (int k = 0; k < 4; ++k)
            v += timef[(size_t)node * 4 + k] * W_time[k * 16 + j];
    }
    h0[t] = v;
}

__global__ void zero_kernel(float* __restrict__ p, int n)
{
    int i = blockIdx.x * blockDim.x + threadIdx.x;
    int stride = gridDim.x * blockDim.x;
    for (; i < n; i += stride) p[i] = 0.0f;
}

// ---------------------------------------------------------------------------
// one wave32 per edge: lane loads float4 of h[src], atomically adds to msg[dst]
// L2-resident atomics (entire msg buffer fits in the 192MB L2)
// ---------------------------------------------------------------------------
__global__ void scatter_kernel(const float* __restrict__ h,
                               const int* __restrict__ src,
                               const int* __restrict__ dst,
                               float* __restrict__ msg,
                               float* __restrict__ deg, int nEdges)
{
    int wid = (blockIdx.x * blockDim.x + threadIdx.x) >> 5;
    int lane = threadIdx.x & 31;
    if (wid >= nEdges) return;
    int s = src[wid];
    int d = dst[wid];
    const float4 v = ((const float4*)(h + (size_t)s * HID))[lane];
    float* md = msg + (size_t)d * HID + lane * 4;
    atomicAdd(md + 0, v.x);
    atomicAdd(md + 1, v.y);
    atomicAdd(md + 2, v.z);
    atomicAdd(md + 3, v.w);
    if (lane == 0) atomicAdd(&deg[d], 1.0f);
}

__global__ void invdeg_kernel(float* __restrict__ deg, int n)
{
    int i = blockIdx.x * blockDim.x + threadIdx.x;
    if (i < n) deg[i] = 1.0f / fmaxf(deg[i], 1.0f);
}

// ---------------------------------------------------------------------------
// out = relu( (msg*invdeg) @ Wl + b + h @ Wr )
// One wave handles 4 row-tiles x 1 column-tile (64 rows x 16 cols of output):
// per K-step the shared B fragments (Wl/Wr columns) are loaded once and reused
// across 4 M-tiles -> 8 f32 WMMAs per 12 loads.
// A layout (16x4 f32): lane = M (mod16); lanes0-15 K={k0,k0+1}, lanes16-31 K={k0+2,k0+3}
// B layout (4x16 f32): VGPR0 = row k0 / k0+2, VGPR1 = row k0+1 / k0+3; N = lane&15
// C/D layout: VGPR j -> M = j (lanes0-15) or j+8 (lanes16-31), N = lane&15
// ---------------------------------------------------------------------------
__global__ void __launch_bounds__(256)
sage_gemm_kernel(const float* __restrict__ msg, const float* __restrict__ invdeg,
                 const float* __restrict__ h, const float* __restrict__ Wl,
                 const float* __restrict__ bias, const float* __restrict__ Wr,
                 float* __restrict__ out, int nTiles)
{
    const int lane  = threadIdx.x & 31;
    const int tile  = blockIdx.x * 8 + (threadIdx.x >> 5); // group*8 + tileN
    const int group = tile >> 3;       // 4 consecutive M row-tiles
    const int tileN = tile & 7;        // which 16-col slab of HID
    const bool hi   = lane >= 16;
    const int cn    = tileN * 16 + (lane & 15);

    int rowT[4];
    const float* mrow[4];
    const float* hrow[4];
    float idg[4];
    #pragma unroll
    for (int m = 0; m < 4; ++m) {
        int rt = group * 4 + m;
        rowT[m] = rt;
        int rc = rt < nTiles ? rt : nTiles - 1;   // clamp loads, guard stores
        int r  = rc * 16 + (lane & 15);
        idg[m]  = invdeg[r];
        mrow[m] = msg + (size_t)r * HID;
        hrow[m] = h   + (size_t)r * HID;
    }

    const float bv = bias[cn];
    v8f acc[4];
    #pragma unroll
    for (int m = 0; m < 4; ++m)
        acc[m] = (v8f){ bv, bv, bv, bv, bv, bv, bv, bv };

#if HAS_WMMA_F32X4
    #pragma unroll 2
    for (int k0 = 0; k0 < HID; k0 += 4) {
        const int ka = k0 + (hi ? 2 : 0);
        v2f b_l, b_r;
        b_l.x = Wl[ka * HID + cn]; b_l.y = Wl[(ka + 1) * HID + cn];
        b_r.x = Wr[ka * HID + cn]; b_r.y = Wr[(ka + 1) * HID + cn];
        #pragma unroll
        for (int m = 0; m < 4; ++m) {
            v2f a_agg, a_h;
            a_agg.x = mrow[m][ka] * idg[m];  a_agg.y = mrow[m][ka + 1] * idg[m];
            a_h.x   = hrow[m][ka];           a_h.y   = hrow[m][ka + 1];
            acc[m] = __builtin_amdgcn_wmma_f32_16x16x4_f32(false, a_agg, false, b_l,
                                                           (short)0, acc[m], false, false);
            acc[m] = __builtin_amdgcn_wmma_f32_16x16x4_f32(false, a_h,   false, b_r,
                                                           (short)0, acc[m], false, false);
        }
    }
#else
    // Scalar fallback (host parse pass / toolchains without the f32 WMMA builtin).
    // Computes the same values directly in the C/D lane layout.
    #pragma unroll
    for (int m = 0; m < 4; ++m) {
        int rc = rowT[m] < nTiles ? rowT[m] : nTiles - 1;
        #pragma unroll
        for (int j = 0; j < 8; ++j) {
            int grow = rc * 16 + (hi ? 8 : 0) + j;
            const float* mr = msg + (size_t)grow * HID;
            const float* hr = h   + (size_t)grow * HID;
            float id = invdeg[grow];
            float s = bv;
            for (int k = 0; k < HID; ++k)
                s += mr[k] * id * Wl[k * HID + cn] + hr[k] * Wr[k * HID + cn];
            acc[m][j] = s;
        }
    }
#endif

    #pragma unroll
    for (int m = 0; m < 4; ++m) {
        if (rowT[m] < nTiles) {
            const int outRowBase = rowT[m] * 16 + (hi ? 8 : 0);
            #pragma unroll
            for (int j = 0; j < 8; ++j) {
                float v = acc[m][j];
                out[(size_t)(outRowBase + j) * HID + cn] = v > 0.0f ? v : 0.0f;
            }
        }
    }
}

// ---------------------------------------------------------------------------
// out[i] = sigmoid(dot(h[i,:], Wc) + bc) ; one wave per node, shuffle reduce
// ---------------------------------------------------------------------------
__global__ void classifier_kernel(const float* __restrict__ h,
                                  const float* __restrict__ Wc,
                                  const float* __restrict__ bc,
                                  float* __restrict__ out, int n)
{
    int wid = (blockIdx.x * blockDim.x + threadIdx.x) >> 5;
    int lane = threadIdx.x & 31;
    if (wid >= n) return;
    const float4 hv = ((const float4*)(h + (size_t)wid * HID))[lane];
    const float4 wv = ((const float4*)Wc)[lane];
    float s = hv.x * wv.x + hv.y * wv.y + hv.z * wv.z + hv.w * wv.w;
    #pragma unroll
    for (int off = 16; off > 0; off >>= 1) s += __shfl_down(s, off, 32);
    if (lane == 0) out[wid] = 1.0f / (1.0f + expf(-(s + bc[0])));
}

// ---------------------------------------------------------------------------
extern "C" void kernel_launch(void* const* d_in, const int* in_sizes, int n_in,
                              void* d_out, int out_size, void* d_ws, size_t ws_size,
                              hipStream_t stream)
{
    const float* x         = (const float*)d_in[0];
    const int*   edge      = (const int*)  d_in[1];
    const int*   user_ids  = (const int*)  d_in[2];
    const int*   locations = (const int*)  d_in[3];
    const float* timef     = (const float*)d_in[4];
    const float* uet       = (const float*)d_in[5];
    const float* let_      = (const float*)d_in[6];
    const float* W_time    = (const float*)d_in[7];
    const float* b_time    = (const float*)d_in[8];
    const float* W1l       = (const float*)d_in[9];
    const float* b1        = (const float*)d_in[10];
    const float* W1r       = (const float*)d_in[11];
    const float* W2l       = (const float*)d_in[12];
    const float* b2        = (const float*)d_in[13];
    const float* W2r       = (const float*)d_in[14];
    const float* Wc        = (const float*)d_in[15];
    const float* bc        = (const float*)d_in[16];
    float* out = (float*)d_out;

    const int N = in_sizes[2];       // 100000 (user_ids count)
    const int E = in_sizes[1] / 2;   // 1600000

    const size_t NH = (size_t)N * HID;
    float* h0  = (float*)d_ws;       // [N,128]
    float* msg = h0  + NH;           // [N,128]
    float* hB  = msg + NH;           // [N,128]
    float* deg = hB  + NH;           // [N]

    const int* src = edge;
    const int* dst = edge + E;

    const int nTiles     = N / 16;                 // 6250 (N divisible by 16)
    const int nhBlocks   = (int)((NH + 255) / 256);
    const int degBlocks  = (N + 255) / 256;
    const int scatBlocks = (E + 7) / 8;            // 8 waves / 256-thread block
    const int gemmBlocks = (nTiles + 3) / 4;       // 1 group (4 M-tiles) x 8 N-tiles per block
    const int clsBlocks  = (N + 7) / 8;

    // Phase A: feature assembly -> h0
    encode_kernel<<<nhBlocks, 256, 0, stream>>>(x, user_ids, locations, timef,
                                                uet, let_, W_time, b_time, h0, N);

    // Layer 1
    zero_kernel<<<nhBlocks, 256, 0, stream>>>(msg, (int)NH);
    zero_kernel<<<degBlocks, 256, 0, stream>>>(deg, N);
    scatter_kernel<<<scatBlocks, 256, 0, stream>>>(h0, src, dst, msg, deg, E);
    invdeg_kernel<<<degBlocks, 256, 0, stream>>>(deg, N);
    sage_gemm_kernel<<<gemmBlocks, 256, 0, stream>>>(msg, deg, h0, W1l, b1, W1r, hB, nTiles);

    // Layer 2
    zero_kernel<<<nhBlocks, 256, 0, stream>>>(msg, (int)NH);
    zero_kernel<<<degBlocks, 256, 0, stream>>>(deg, N);
    scatter_kernel<<<scatBlocks, 256, 0, stream>>>(hB, src, dst, msg, deg, E);
    invdeg_kernel<<<degBlocks, 256, 0, stream>>>(deg, N);
    sage_gemm_kernel<<<gemmBlocks, 256, 0, stream>>>(msg, deg, hB, W2l, b2, W2r, h0, nTiles);

    // Classifier
    classifier_kernel<<<clsBlocks, 256, 0, stream>>>(h0, Wc, bc, out, N);
}